// MultiHeadSelfAttention_82334523064542
// MI455X (gfx1250) — compile-verified
//
#include <hip/hip_runtime.h>
#include <math.h>

typedef __attribute__((ext_vector_type(2))) float v2f;
typedef __attribute__((ext_vector_type(8))) float v8f;

#define S_LEN 2048
#define DMODEL 128
#define HDIM 32
#define ROWS (4 * 2048) /* B*S */

// D = A(16x4 f32) * B(4x16 f32) + C(16x16 f32)
__device__ inline v8f wmma4(v2f a, v2f b, v8f c) {
  return __builtin_amdgcn_wmma_f32_16x16x4_f32(false, a, false, b, (short)0, c,
                                               false, false);
}

__device__ inline void lds_fence() { asm volatile("s_wait_dscnt 0" ::: "memory"); }

__device__ inline float rowmax16(float v) {
  v = fmaxf(v, __shfl_xor(v, 1, 32));
  v = fmaxf(v, __shfl_xor(v, 2, 32));
  v = fmaxf(v, __shfl_xor(v, 4, 32));
  v = fmaxf(v, __shfl_xor(v, 8, 32));
  return v;
}
__device__ inline float rowsum16(float v) {
  v += __shfl_xor(v, 1, 32);
  v += __shfl_xor(v, 2, 32);
  v += __shfl_xor(v, 4, 32);
  v += __shfl_xor(v, 8, 32);
  return v;
}

// ---------------- LayerNorm: one 128-thread block per row ----------------
__global__ __launch_bounds__(128) void ln_kernel(const float* __restrict__ x,
                                                 const float* __restrict__ gamma,
                                                 const float* __restrict__ beta,
                                                 float* __restrict__ xn) {
  const int row = blockIdx.x;
  const int t = threadIdx.x;
  __shared__ float red[128];
  const float v = x[(size_t)row * DMODEL + t];
  red[t] = v;
  __syncthreads();
  for (int off = 64; off > 0; off >>= 1) {
    if (t < off) red[t] += red[t + off];
    __syncthreads();
  }
  const float mu = red[0] * (1.0f / DMODEL);
  __syncthreads();
  const float d = v - mu;
  red[t] = d * d;
  __syncthreads();
  for (int off = 64; off > 0; off >>= 1) {
    if (t < off) red[t] += red[t + off];
    __syncthreads();
  }
  const float var = red[0] * (1.0f / DMODEL);
  xn[(size_t)row * DMODEL + t] = d * rsqrtf(var + 1e-6f) * gamma[t] + beta[t];
}

// ------------- GEMM: out[M,N] = A[M,K] @ W[K,N] + bias (+resid) ----------
// one wave per 16x16 output tile, fp32 WMMA, K multiple of 4
__global__ __launch_bounds__(32) void gemm_kernel(const float* __restrict__ A,
                                                  const float* __restrict__ W,
                                                  const float* __restrict__ bias,
                                                  const float* __restrict__ resid,
                                                  float* __restrict__ out,
                                                  int N, int K) {
  const int lane = threadIdx.x & 31;
  const int half = lane >> 4;
  const int l16 = lane & 15;
  const int m0 = blockIdx.x * 16;
  const int n0 = blockIdx.y * 16;

  v8f acc = {0, 0, 0, 0, 0, 0, 0, 0};
  const float* arow = A + (size_t)(m0 + l16) * K + 2 * half;
  for (int k0 = 0; k0 < K; k0 += 4) {
    v2f a;
    a.x = arow[k0];
    a.y = arow[k0 + 1];
    const int kr = k0 + 2 * half;
    v2f b;
    b.x = W[(size_t)kr * N + n0 + l16];
    b.y = W[(size_t)(kr + 1) * N + n0 + l16];
    acc = wmma4(a, b, acc);
  }
  const float bn = bias[n0 + l16];
#pragma unroll
  for (int i = 0; i < 8; ++i) {
    const int m = i + 8 * half;
    const size_t idx = (size_t)(m0 + m) * N + n0 + l16;
    float r = acc[i] + bn;
    if (resid) r += resid[idx];
    out[idx] = r;
  }
}

// ----------------- Flash attention, one wave per (b,h,qtile) -------------
__global__ __launch_bounds__(32) void attn_kernel(const float* __restrict__ Q,
                                                  const float* __restrict__ K,
                                                  const float* __restrict__ V,
                                                  float* __restrict__ CTX) {
  const int lane = threadIdx.x & 31;
  const int half = lane >> 4;
  const int l16 = lane & 15;
  const int qt = blockIdx.x;  // 0..127
  const int bh = blockIdx.y;  // 0..15
  const int b = bh >> 2, h = bh & 3;
  const size_t bhBase = (size_t)b * S_LEN * DMODEL + (size_t)h * HDIM;
  const int q0 = qt * 16;

  // Q tile as 8 A-fragments (Dh = 32 = 8 k-slices of 4)
  v2f aQ[8];
  {
    const float* qrow = Q + bhBase + (size_t)(q0 + l16) * DMODEL + 2 * half;
#pragma unroll
    for (int f = 0; f < 8; ++f) {
      aQ[f].x = qrow[4 * f];
      aQ[f].y = qrow[4 * f + 1];
    }
  }

  float m_i[8], l_i[8];
#pragma unroll
  for (int i = 0; i < 8; ++i) {
    m_i[i] = -1e30f;
    l_i[i] = 0.0f;
  }
  v8f o0 = {0, 0, 0, 0, 0, 0, 0, 0};
  v8f o1 = {0, 0, 0, 0, 0, 0, 0, 0};

  __shared__ float Pl[16][17];
  const float scale = 0.17677669529663687f;  // 1/sqrt(32)

  for (int kt = 0; kt < S_LEN / 16; ++kt) {
    const int k0 = kt * 16;

    // K tile as 8 B-fragments: B[k][n] = K[key = k0+n][dh = k]
    v2f bK[8];
    const float* krow = K + bhBase + (size_t)(k0 + l16) * DMODEL + 2 * half;
#pragma unroll
    for (int f = 0; f < 8; ++f) {
      bK[f].x = krow[4 * f];
      bK[f].y = krow[4 * f + 1];
    }

    v8f sc = {0, 0, 0, 0, 0, 0, 0, 0};
#pragma unroll
    for (int f = 0; f < 8; ++f) sc = wmma4(aQ[f], bK[f], sc);

    // online softmax on the 16x16 score tile (row m = i + 8*half, col = l16)
    float p[8];
#pragma unroll
    for (int i = 0; i < 8; ++i) {
      const float sv = sc[i] * scale;
      const float rm = rowmax16(sv);
      const float nm = fmaxf(m_i[i], rm);
      const float corr = __expf(m_i[i] - nm);
      m_i[i] = nm;
      p[i] = __expf(sv - nm);
      const float rs = rowsum16(p[i]);
      l_i[i] = l_i[i] * corr + rs;
      o0[i] *= corr;
      o1[i] *= corr;
    }

    // C-layout -> A-layout via LDS (single wave, fence with s_wait_dscnt)
#pragma unroll
    for (int i = 0; i < 8; ++i) Pl[i + 8 * half][l16] = p[i];
    lds_fence();
    v2f aP[4];
#pragma unroll
    for (int f = 0; f < 4; ++f) {
      aP[f].x = Pl[l16][4 * f + 2 * half];
      aP[f].y = Pl[l16][4 * f + 2 * half + 1];
    }
    lds_fence();

    // O += P(16x16) * V(16x32): two N-halves of Dh
    const float* vbase = V + bhBase;
#pragma unroll
    for (int f = 0; f < 4; ++f) {
      const int kr = k0 + 4 * f + 2 * half;
      v2f b0, b1;
      b0.x = vbase[(size_t)kr * DMODEL + l16];
      b0.y = vbase[(size_t)(kr + 1) * DMODEL + l16];
      b1.x = vbase[(size_t)kr * DMODEL + 16 + l16];
      b1.y = vbase[(size_t)(kr + 1) * DMODEL + 16 + l16];
      o0 = wmma4(aP[f], b0, o0);
      o1 = wmma4(aP[f], b1, o1);
    }
  }

#pragma unroll
  for (int i = 0; i < 8; ++i) {
    const float inv = 1.0f / l_i[i];
    const int m = i + 8 * half;
    const size_t row = (size_t)(b * S_LEN + q0 + m) * DMODEL + (size_t)h * HDIM;
    CTX[row + l16] = o0[i] * inv;
    CTX[row + 16 + l16] = o1[i] * inv;
  }
}

extern "C" void kernel_launch(void* const* d_in, const int* in_sizes, int n_in,
                              void* d_out, int out_size, void* d_ws, size_t ws_size,
                              hipStream_t stream) {
  (void)in_sizes; (void)n_in; (void)out_size; (void)ws_size;
  const float* x = (const float*)d_in[0];
  const float* Wq = (const float*)d_in[1];
  const float* bq = (const float*)d_in[2];
  const float* Wk = (const float*)d_in[3];
  const float* bk = (const float*)d_in[4];
  const float* Wv = (const float*)d_in[5];
  const float* bv = (const float*)d_in[6];
  const float* gamma = (const float*)d_in[7];
  const float* beta = (const float*)d_in[8];
  const float* Wo = (const float*)d_in[9];
  const float* bo = (const float*)d_in[10];
  float* out = (float*)d_out;

  const size_t mat = (size_t)ROWS * DMODEL;  // 1M floats
  float* xn = (float*)d_ws;
  float* q = xn + mat;
  float* k = q + mat;
  float* v = k + mat;
  float* ctx = v + mat;

  ln_kernel<<<ROWS, 128, 0, stream>>>(x, gamma, beta, xn);

  dim3 gg(ROWS / 16, DMODEL / 16);
  gemm_kernel<<<gg, 32, 0, stream>>>(xn, Wq, bq, nullptr, q, DMODEL, DMODEL);
  gemm_kernel<<<gg, 32, 0, stream>>>(xn, Wk, bk, nullptr, k, DMODEL, DMODEL);
  gemm_kernel<<<gg, 32, 0, stream>>>(xn, Wv, bv, nullptr, v, DMODEL, DMODEL);

  dim3 ga(S_LEN / 16, 16);
  attn_kernel<<<ga, 32, 0, stream>>>(q, k, v, ctx);

  gemm_kernel<<<gg, 32, 0, stream>>>(ctx, Wo, bo, x, out, DMODEL, DMODEL);
}